// Noise_9216999817324
// MI455X (gfx1250) — compile-verified
//
#include <hip/hip_runtime.h>
#include <stdint.h>

// ext_vector types so nontemporal builtins accept them and lower to b128 VMEM.
typedef float v4f __attribute__((ext_vector_type(4)));
typedef int   v4i __attribute__((ext_vector_type(4)));

#define NOISE_MASK 0x7fffffffu

// Finalize a lattice-corner hash. `n` is the wrapped sum ix*1619 + iy*31337 + seed*1013.
// All arithmetic in uint32 (wraps identically to JAX int32 two's complement).
__device__ __forceinline__ float corner_from_sum(uint32_t n) {
    n &= NOISE_MASK;                 // non-negative, so >> is sign-agnostic
    n = (n >> 13) ^ n;
    n = (n * (n * n * 60493u + 19990303u) + 1376312589u) & NOISE_MASK;
    // n / 2^30 : reciprocal multiply is exact for a power of two.
    return 1.0f - (float)n * (1.0f / 1073741824.0f);
}

__device__ __forceinline__ float value_noise_point(float x, float y, int seed) {
    float x0f = floorf(x);
    float y0f = floorf(y);
    float fx  = x - x0f;
    float fy  = y - y0f;
    int ix0 = (int)x0f;
    int iy0 = (int)y0f;

    // CSE: hash base for (ix0, seed); ix1 adds one generator step, same for y.
    uint32_t hx0 = (uint32_t)ix0 * 1619u + (uint32_t)seed * 1013u;
    uint32_t hx1 = hx0 + 1619u;
    uint32_t hy0 = (uint32_t)iy0 * 31337u;
    uint32_t hy1 = hy0 + 31337u;

    float c00 = corner_from_sum(hx0 + hy0);
    float c10 = corner_from_sum(hx1 + hy0);
    float c01 = corner_from_sum(hx0 + hy1);
    float c11 = corner_from_sum(hx1 + hy1);

    float top = c00 + fx * (c10 - c00);
    float bot = c01 + fx * (c11 - c01);
    return top + fy * (bot - top);
}

// 4 points per thread: 2x b128 position loads, 1x b128 seed load, 1x b128 store.
// Streaming workload (134 MB, zero reuse) -> non-temporal hints keep L2 clean.
__global__ __launch_bounds__(256) void noise_value2d_kernel(
    const v4f* __restrict__ pos2,   // N/2 float4s; each holds 2 (x,y) points
    const v4i* __restrict__ seed4,  // N/4 int4s
    v4f* __restrict__ out4,         // N/4 float4s
    int n_quads)
{
    int g = blockIdx.x * blockDim.x + threadIdx.x;
    if (g >= n_quads) return;

    v4f p01 = __builtin_nontemporal_load(&pos2[2 * g + 0]); // points 0,1
    v4f p23 = __builtin_nontemporal_load(&pos2[2 * g + 1]); // points 2,3
    v4i s   = __builtin_nontemporal_load(&seed4[g]);

    v4f r;
    r.x = value_noise_point(p01.x, p01.y, s.x);
    r.y = value_noise_point(p01.z, p01.w, s.y);
    r.z = value_noise_point(p23.x, p23.y, s.z);
    r.w = value_noise_point(p23.z, p23.w, s.w);

    __builtin_nontemporal_store(r, &out4[g]);
}

extern "C" void kernel_launch(void* const* d_in, const int* in_sizes, int n_in,
                              void* d_out, int out_size, void* d_ws, size_t ws_size,
                              hipStream_t stream) {
    (void)n_in; (void)d_ws; (void)ws_size;

    const v4f* pos2  = (const v4f*)d_in[0];   // position: (N,2) float32, interleaved x,y
    const v4i* seed4 = (const v4i*)d_in[1];   // seed: (N,) int32
    v4f*       out4  = (v4f*)d_out;           // noise: (N,) float32

    int n_points = in_sizes[1];               // seed count == N
    int n_quads  = n_points >> 2;             // N = 8388608, divisible by 4

    const int block = 256;                    // 8 wave32s per block
    int grid = (n_quads + block - 1) / block;

    noise_value2d_kernel<<<grid, block, 0, stream>>>(pos2, seed4, out4, n_quads);
}